// PointNetPlusPlus_33964601376854
// MI455X (gfx1250) — compile-verified
//
#include <hip/hip_runtime.h>
#include <hip/hip_bf16.h>
#include <cstdint>
#include <cstddef>

typedef float v2f __attribute__((ext_vector_type(2)));
typedef float v8f __attribute__((ext_vector_type(8)));

#define BN_EPS 1e-5f
#define NB 16   // batch
#define NPTS 8192

// ---------------------------------------------------------------------------
// prep: xyz (B,3,N) -> xyzT (B,N,3); points = concat(xyz,features) -> (B,N,8)
// ---------------------------------------------------------------------------
__global__ void prep_kernel(const float* __restrict__ xyz,
                            const float* __restrict__ feat,
                            float* __restrict__ xyzT,
                            float* __restrict__ pts, int N)
{
  int i = blockIdx.x * blockDim.x + threadIdx.x;
  if (i >= NB * N) return;
  int b = i / N, n = i % N;
  const float* xb = xyz + (size_t)b * 3 * N;
  const float* fb = feat + (size_t)b * 5 * N;
  float x = xb[n], y = xb[N + n], z = xb[2 * N + n];
  float* o = xyzT + (size_t)i * 3;
  o[0] = x; o[1] = y; o[2] = z;
  float* p = pts + (size_t)i * 8;
  p[0] = x; p[1] = y; p[2] = z;
#pragma unroll
  for (int c = 0; c < 5; ++c) p[3 + c] = fb[(size_t)c * N + n];
}

// ---------------------------------------------------------------------------
// Farthest point sampling: one block per batch, dist[] in LDS.
// Mirrors the lax.scan exactly: sample k = `far` entering iteration k,
// far starts at 0; argmax tie-break = lowest index (jnp.argmax).
// ---------------------------------------------------------------------------
__global__ void fps_kernel(const float* __restrict__ xyzT, int N, int npoint,
                           float* __restrict__ out_xyz)
{
  extern __shared__ float smem[];
  float* dist = smem;                 // N floats
  float* rv = smem + N;               // 256 floats
  int* ri = (int*)(rv + 256);         // 256 ints
  const int b = blockIdx.x, t = threadIdx.x;
  const float* P = xyzT + (size_t)b * N * 3;
  for (int j = t; j < N; j += 256) dist[j] = 1e10f;
  __syncthreads();
  int far = 0;
  for (int it = 0; it < npoint; ++it) {
    float cx = P[far * 3], cy = P[far * 3 + 1], cz = P[far * 3 + 2];
    if (t == 0) {
      float* o = out_xyz + ((size_t)b * npoint + it) * 3;
      o[0] = cx; o[1] = cy; o[2] = cz;
    }
    float bm = -1.0f; int bi = 0;
    for (int j = t; j < N; j += 256) {
      float dx = P[j * 3] - cx, dy = P[j * 3 + 1] - cy, dz = P[j * 3 + 2] - cz;
      float d = dx * dx + dy * dy + dz * dz;
      float nd = fminf(dist[j], d);
      dist[j] = nd;
      if (nd > bm) { bm = nd; bi = j; }   // strict > keeps lowest j per thread
    }
    rv[t] = bm; ri[t] = bi;
    __syncthreads();
    for (int s = 128; s > 0; s >>= 1) {
      if (t < s) {
        if (rv[t + s] > rv[t] || (rv[t + s] == rv[t] && ri[t + s] < ri[t])) {
          rv[t] = rv[t + s]; ri[t] = ri[t + s];
        }
      }
      __syncthreads();
    }
    far = ri[0];
    __syncthreads();
  }
}

// ---------------------------------------------------------------------------
// Ball query + group: first NS indices (ascending) with d<=r2, padded with
// the first hit; writes grouped rows [dx,dy,dz, feat..., zero-pad] (KP cols).
// ---------------------------------------------------------------------------
template <int NS, int CF, int KP>
__global__ void group_kernel(const float* __restrict__ xyzT,
                             const float* __restrict__ pts,
                             const float* __restrict__ nxyz,
                             float* __restrict__ Xin, int N, int NP, float r2)
{
  int gid = blockIdx.x * blockDim.x + threadIdx.x;
  if (gid >= NB * NP) return;
  int b = gid / NP;
  const float* P = xyzT + (size_t)b * N * 3;
  const float* F = pts + (size_t)b * N * CF;
  float cx = nxyz[(size_t)gid * 3 + 0];
  float cy = nxyz[(size_t)gid * 3 + 1];
  float cz = nxyz[(size_t)gid * 3 + 2];
  int sel[NS];
  int cnt = 0;
  for (int j = 0; j < N; ++j) {
    float dx = P[j * 3] - cx, dy = P[j * 3 + 1] - cy, dz = P[j * 3 + 2] - cz;
    float d = dx * dx + dy * dy + dz * dz;
    if (d <= r2) { sel[cnt++] = j; if (cnt == NS) break; }
  }
  int first = (cnt > 0) ? sel[0] : 0;
  for (int s = cnt; s < NS; ++s) sel[s] = first;
  for (int s = 0; s < NS; ++s) {
    int j = sel[s];
    float* row = Xin + ((size_t)gid * NS + s) * KP;
    row[0] = P[j * 3] - cx; row[1] = P[j * 3 + 1] - cy; row[2] = P[j * 3 + 2] - cz;
    for (int c = 0; c < CF; ++c) row[3 + c] = F[(size_t)j * CF + c];
    for (int c = 3 + CF; c < KP; ++c) row[c] = 0.0f;
  }
}

// group_all (SA3): rows = B*NP, cols = [xyz(3), feat(CF), pad]
__global__ void group_all_kernel(const float* __restrict__ nxyz,
                                 const float* __restrict__ pool,
                                 float* __restrict__ Xin, int NP, int CF, int KP)
{
  int i = blockIdx.x * blockDim.x + threadIdx.x;
  if (i >= NB * NP) return;
  float* row = Xin + (size_t)i * KP;
  row[0] = nxyz[(size_t)i * 3 + 0];
  row[1] = nxyz[(size_t)i * 3 + 1];
  row[2] = nxyz[(size_t)i * 3 + 2];
  for (int c = 0; c < CF; ++c) row[3 + c] = pool[(size_t)i * CF + c];
  for (int c = 3 + CF; c < KP; ++c) row[c] = 0.0f;
}

// pad W (Nc x K) -> Wp (Nc x Kp), zero tail: kills all guards in the GEMM loop
__global__ void pad_weights_kernel(const float* __restrict__ W,
                                   float* __restrict__ Wp, int Nc, int K, int Kp)
{
  int i = blockIdx.x * blockDim.x + threadIdx.x;
  if (i >= Nc * Kp) return;
  int n = i / Kp, k = i % Kp;
  Wp[i] = (k < K) ? W[(size_t)n * K + k] : 0.0f;
}

// ---------------------------------------------------------------------------
// WMMA f32 16x16x4 GEMM:  Y(MxNc) = X(MxKp) * Wp^T + bias,  Wp is (Nc x Kp).
// One wave32 computes a 16x(16*NT) strip; A fragment loaded once per K-step
// and reused for NT B-tiles. Software-pipelined one K-step deep: loads for
// step k+4 are issued before the WMMAs of step k so the loadcnt waits overlap
// with matrix math instead of stalling on a fresh load.
// Fragment layouts per ISA §7.12.2:
//   A (16x4 f32): lane = half*16 + m; vgpr0/1 = K (k + 2*half) / (+1)
//   B (4x16 f32): mirrored over columns
//   C/D: vgpr r -> row r + 8*half, col = lane&15
// ---------------------------------------------------------------------------
template <int NT>
__global__ __launch_bounds__(32)
void gemm_bias_wmma(const float* __restrict__ X, const float* __restrict__ Wp,
                    const float* __restrict__ bias, float* __restrict__ Y,
                    int M, int Nc, int Kp)
{
  const int lane = threadIdx.x;
  const int half = lane >> 4;
  const int l16 = lane & 15;
  const int m = (blockIdx.x << 4) + l16;
  const int nbase = blockIdx.y * (NT * 16);
  const float* xrow = X + (size_t)m * Kp + (half << 1);
  const float* wrow[NT];
#pragma unroll
  for (int t = 0; t < NT; ++t)
    wrow[t] = Wp + (size_t)(nbase + t * 16 + l16) * Kp + (half << 1);

  v8f acc[NT];
#pragma unroll
  for (int t = 0; t < NT; ++t) acc[t] = (v8f){};

  // prologue: stage k=0 fragments
  v2f a_cur = *(const v2f*)(xrow);
  v2f b_cur[NT];
#pragma unroll
  for (int t = 0; t < NT; ++t) b_cur[t] = *(const v2f*)(wrow[t]);

  for (int k = 4; k < Kp; k += 4) {
    // issue next-stage loads first ...
    v2f a_nxt = *(const v2f*)(xrow + k);
    v2f b_nxt[NT];
#pragma unroll
    for (int t = 0; t < NT; ++t) b_nxt[t] = *(const v2f*)(wrow[t] + k);
    // ... then do the math on the previous stage (already in flight/resident)
#pragma unroll
    for (int t = 0; t < NT; ++t)
      acc[t] = __builtin_amdgcn_wmma_f32_16x16x4_f32(false, a_cur, false,
                                                     b_cur[t], (short)0,
                                                     acc[t], false, false);
    a_cur = a_nxt;
#pragma unroll
    for (int t = 0; t < NT; ++t) b_cur[t] = b_nxt[t];
  }
  // epilogue: last K-step
#pragma unroll
  for (int t = 0; t < NT; ++t)
    acc[t] = __builtin_amdgcn_wmma_f32_16x16x4_f32(false, a_cur, false,
                                                   b_cur[t], (short)0,
                                                   acc[t], false, false);

#pragma unroll
  for (int t = 0; t < NT; ++t) {
    const int col = nbase + t * 16 + l16;
    const float bv = bias[col];
#pragma unroll
    for (int r = 0; r < 8; ++r) {
      const int row = (blockIdx.x << 4) + r + (half << 3);
      Y[(size_t)row * Nc + col] = acc[t][r] + bv;
    }
  }
}

// per-channel mean / inv-std over all M rows (population var, like jnp.var)
__global__ void col_stats_kernel(const float* __restrict__ Y, int M, int C,
                                 float* __restrict__ mean, float* __restrict__ invstd)
{
  __shared__ double sh[256], shq[256];
  const int c = blockIdx.x, t = threadIdx.x;
  double s = 0.0, sq = 0.0;
  for (int r = t; r < M; r += 256) {
    float v = Y[(size_t)r * C + c];
    s += v; sq += (double)v * v;
  }
  sh[t] = s; shq[t] = sq;
  __syncthreads();
  for (int k = 128; k > 0; k >>= 1) {
    if (t < k) { sh[t] += sh[t + k]; shq[t] += shq[t + k]; }
    __syncthreads();
  }
  if (t == 0) {
    double mu = sh[0] / M;
    double var = shq[0] / M - mu * mu;
    mean[c] = (float)mu;
    invstd[c] = rsqrtf((float)var + BN_EPS);
  }
}

__global__ void bnrelu_kernel(float* __restrict__ Y, size_t total, int C,
                              const float* __restrict__ mean,
                              const float* __restrict__ invstd,
                              const float* __restrict__ g,
                              const float* __restrict__ be)
{
  size_t i = (size_t)blockIdx.x * blockDim.x + threadIdx.x;
  if (i >= total) return;
  int c = (int)(i % C);
  float v = (Y[i] - mean[c]) * invstd[c] * g[c] + be[c];
  Y[i] = fmaxf(v, 0.0f);
}

// max over the NS (neighbor) dimension: in rows = rows_out*NS, out rows_out x C
__global__ void maxpool_kernel(const float* __restrict__ Y, float* __restrict__ out,
                               int rows_out, int NS, int C)
{
  int i = blockIdx.x * blockDim.x + threadIdx.x;
  if (i >= rows_out * C) return;
  int row = i / C, c = i % C;
  const float* base = Y + ((size_t)row * NS) * C + c;
  float m = base[0];
  for (int s = 1; s < NS; ++s) m = fmaxf(m, base[(size_t)s * C]);
  out[(size_t)row * C + c] = m;
}

// ---------------------------------------------------------------------------
// host side
// ---------------------------------------------------------------------------
static void run_layer(const float* X, int M, int K, int Kp, int Cout,
                      const float* W, const float* b, const float* g,
                      const float* be, float* Wp, float* Y, float* stats,
                      hipStream_t s)
{
  pad_weights_kernel<<<(Cout * Kp + 255) / 256, 256, 0, s>>>(W, Wp, Cout, K, Kp);
  dim3 grid(M / 16, Cout / 64);
  gemm_bias_wmma<4><<<grid, 32, 0, s>>>(X, Wp, b, Y, M, Cout, Kp);
  col_stats_kernel<<<Cout, 256, 0, s>>>(Y, M, Cout, stats, stats + Cout);
  size_t total = (size_t)M * Cout;
  bnrelu_kernel<<<(unsigned)((total + 255) / 256), 256, 0, s>>>(
      Y, total, Cout, stats, stats + Cout, g, be);
}

extern "C" void kernel_launch(void* const* d_in, const int* in_sizes, int n_in,
                              void* d_out, int out_size, void* d_ws, size_t ws_size,
                              hipStream_t stream)
{
  (void)in_sizes; (void)n_in; (void)out_size; (void)ws_size;
  const float* xyz  = (const float*)d_in[0];
  const float* feat = (const float*)d_in[1];
  const float* P[36];
  for (int i = 0; i < 36; ++i) P[i] = (const float*)d_in[2 + i];
  // P layout: per layer (W, b, g, be); sa1 = P[0..11], sa2 = P[12..23], sa3 = P[24..35]

  char* w = (char*)d_ws;
  size_t off = 0;
  auto arena = [&](size_t bytes) -> void* {
    void* p = w + off;
    off += (bytes + 255) & ~(size_t)255;
    return p;
  };
  float* xyzT  = (float*)arena((size_t)NB * NPTS * 3 * 4);
  float* pts0  = (float*)arena((size_t)NB * NPTS * 8 * 4);
  float* nx1   = (float*)arena((size_t)NB * 512 * 3 * 4);
  float* pool1 = (float*)arena((size_t)NB * 512 * 128 * 4);
  float* nx2   = (float*)arena((size_t)NB * 128 * 3 * 4);
  float* pool2 = (float*)arena((size_t)NB * 128 * 256 * 4);
  float* stats = (float*)arena(2 * 1024 * 4);
  float* Wp    = (float*)arena((size_t)1024 * 512 * 4);   // max Nc*Kp
  float* bufA  = (float*)arena((size_t)33554432 * 4);     // 128 MB
  float* bufB  = (float*)arena((size_t)33554432 * 4);     // 128 MB

  // ---- prep ----
  prep_kernel<<<(NB * NPTS + 255) / 256, 256, 0, stream>>>(xyz, feat, xyzT, pts0, NPTS);

  // ---- SA1: FPS 512, ball r=0.2 ns=32, MLP 11->64->64->128 ----
  fps_kernel<<<NB, 256, NPTS * 4 + 256 * 8, stream>>>(xyzT, NPTS, 512, nx1);
  group_kernel<32, 8, 12><<<(NB * 512 + 255) / 256, 256, 0, stream>>>(
      xyzT, pts0, nx1, bufB, NPTS, 512, 0.04f);
  const int M1 = NB * 512 * 32;                      // 262144
  run_layer(bufB, M1, 11, 12, 64, P[0], P[1], P[2], P[3], Wp, bufA, stats, stream);
  run_layer(bufA, M1, 64, 64, 64, P[4], P[5], P[6], P[7], Wp, bufB, stats, stream);
  run_layer(bufB, M1, 64, 64, 128, P[8], P[9], P[10], P[11], Wp, bufA, stats, stream);
  maxpool_kernel<<<(NB * 512 * 128 + 255) / 256, 256, 0, stream>>>(
      bufA, pool1, NB * 512, 32, 128);

  // ---- SA2: FPS 128 over 512 pts, ball r=0.4 ns=64, MLP 131->128->128->256 ----
  fps_kernel<<<NB, 256, 512 * 4 + 256 * 8, stream>>>(nx1, 512, 128, nx2);
  group_kernel<64, 128, 132><<<(NB * 128 + 255) / 256, 256, 0, stream>>>(
      nx1, pool1, nx2, bufB, 512, 128, 0.16f);
  const int M2 = NB * 128 * 64;                      // 131072
  run_layer(bufB, M2, 131, 132, 128, P[12], P[13], P[14], P[15], Wp, bufA, stats, stream);
  run_layer(bufA, M2, 128, 128, 128, P[16], P[17], P[18], P[19], Wp, bufB, stats, stream);
  run_layer(bufB, M2, 128, 128, 256, P[20], P[21], P[22], P[23], Wp, bufA, stats, stream);
  maxpool_kernel<<<(NB * 128 * 256 + 255) / 256, 256, 0, stream>>>(
      bufA, pool2, NB * 128, 64, 256);

  // ---- SA3: group-all, MLP 259->256->512->1024, max over 128 ----
  group_all_kernel<<<(NB * 128 + 255) / 256, 256, 0, stream>>>(
      nx2, pool2, bufB, 128, 256, 260);
  const int M3 = NB * 128;                           // 2048
  run_layer(bufB, M3, 259, 260, 256, P[24], P[25], P[26], P[27], Wp, bufA, stats, stream);
  run_layer(bufA, M3, 256, 256, 512, P[28], P[29], P[30], P[31], Wp, bufB, stats, stream);
  run_layer(bufB, M3, 512, 512, 1024, P[32], P[33], P[34], P[35], Wp, bufA, stats, stream);
  maxpool_kernel<<<(NB * 1024 + 255) / 256, 256, 0, stream>>>(
      bufA, (float*)d_out, NB, 128, 1024);
}